// Point2Voxel_48584670053111
// MI455X (gfx1250) — compile-verified
//
#include <hip/hip_runtime.h>

// Problem constants (match reference)
#define GRID_N   64
#define UNIT_F   0.05f
#define HALF_F   1.6f          // 0.5 * 64 * 0.05
#define BATCHES  8
#define VOXELS   (64 * 64 * 64)          // 262144 per batch
#define CH       32                      // feature channels == wave32 lanes

// clang ext-vector: valid operand for __builtin_nontemporal_store (128-bit)
typedef float __attribute__((ext_vector_type(4))) v4f;

// ---------------------------------------------------------------------------
// Kernel 1: zero the accumulator region (voxel_feats + voxel_counts + inst_flag)
// via the CDNA5 async LDS->global DMA path: zero a 4KB LDS staging buffer once,
// then each lane streams GLOBAL_STORE_ASYNC_FROM_LDS_B128 (ASYNCcnt-tracked)
// in a grid-stride loop; drain with s_wait_asynccnt 0.
// ---------------------------------------------------------------------------
__global__ void zero_accum(float* __restrict__ base, long n4) {
    __shared__ v4f zbuf[256];                  // 4KB, one 16B chunk per thread
    zbuf[threadIdx.x] = (v4f){0.f, 0.f, 0.f, 0.f};
    __syncthreads();

    // Raw 32-bit LDS byte offset of this thread's staging chunk
    // (low 32 bits of its generic address).
    unsigned lds_off = (unsigned)(size_t)&zbuf[threadIdx.x];

    long i      = (long)blockIdx.x * blockDim.x + threadIdx.x;
    long stride = (long)gridDim.x * blockDim.x;
    for (; i < n4; i += stride) {
        float* gptr = base + i * 4;            // 16B-aligned destination
        asm volatile(
            "global_store_async_from_lds_b128 %0, %1, off"
            :
            : "v"(gptr), "v"(lds_off)
            : "memory");
    }
    asm volatile("s_wait_asynccnt 0x0" ::: "memory");
}

// ---------------------------------------------------------------------------
// Kernel 2: one wave32 per point, lane = channel.
//  - all lanes load the 3 pc coords (broadcast addresses), compute valid/lin
//  - lane 0 writes masked_pc, count atomic, inst flag (same-value race: benign)
//  - every lane does one hardware f32 atomic add on its channel
// ---------------------------------------------------------------------------
__global__ void scatter_points(const float* __restrict__ pc,
                               const float* __restrict__ feat,
                               const int*   __restrict__ bid,
                               float* __restrict__ masked,
                               float* __restrict__ sums,
                               float* __restrict__ counts,
                               float* __restrict__ inst,
                               int N) {
    int tid  = blockIdx.x * blockDim.x + threadIdx.x;
    int p    = tid >> 5;           // point index (one wave per point)
    int lane = tid & 31;           // channel index
    if (p >= N) return;

    float x = pc[3 * p + 0];
    float y = pc[3 * p + 1];
    float z = pc[3 * p + 2];
    bool valid = (fabsf(x) <= HALF_F) && (fabsf(y) <= HALF_F) && (fabsf(z) <= HALF_F);

    if (lane == 0) {
        masked[3 * p + 0] = valid ? x : 0.0f;
        masked[3 * p + 1] = valid ? y : 0.0f;
        masked[3 * p + 2] = valid ? z : 0.0f;
    }
    if (!valid) return;

    // idx = int32((pc + HALF) / UNIT), clipped to [0,63]. For valid points the
    // argument is non-negative so C truncation == jnp astype(int32).
    int ix = (int)((x + HALF_F) / UNIT_F);
    int iy = (int)((y + HALF_F) / UNIT_F);
    int iz = (int)((z + HALF_F) / UNIT_F);
    ix = min(max(ix, 0), GRID_N - 1);
    iy = min(max(iy, 0), GRID_N - 1);
    iz = min(max(iz, 0), GRID_N - 1);

    int  b   = bid[p];
    long lin = (long)b * VOXELS + (long)ix * (GRID_N * GRID_N) + iy * GRID_N + iz;

    // Streaming single-use read of this point's feature row (coalesced 128B/wave)
    float f = __builtin_nontemporal_load(&feat[(long)p * CH + lane]);

    // Hardware FP32 global atomic (global_atomic_add_f32), consecutive lanes ->
    // consecutive DWORDs in the voxel's 128B feature row.
    unsafeAtomicAdd(&sums[lin * CH + lane], f);

    if (lane == 0) {
        unsafeAtomicAdd(&counts[lin], 1.0f);
        inst[b] = 1.0f;   // all writers store identical value
    }
}

// ---------------------------------------------------------------------------
// Kernel 3: one wave32 per voxel; uniform count, divide the 32-channel row.
// Matches reference: sums / max(counts, 1.0)  (true division).
// ---------------------------------------------------------------------------
__global__ void finalize_voxels(float* __restrict__ feats,
                                const float* __restrict__ counts,
                                long BV) {
    long tid  = (long)blockIdx.x * blockDim.x + threadIdx.x;
    long v    = tid >> 5;
    int  lane = (int)(tid & 31);
    if (v >= BV) return;
    float c = counts[v];              // uniform across the wave
    float d = fmaxf(c, 1.0f);
    feats[v * CH + lane] = feats[v * CH + lane] / d;
}

// ---------------------------------------------------------------------------
extern "C" void kernel_launch(void* const* d_in, const int* in_sizes, int n_in,
                              void* d_out, int out_size, void* d_ws, size_t ws_size,
                              hipStream_t stream) {
    const float* pc   = (const float*)d_in[0];   // (N,3) f32
    const float* feat = (const float*)d_in[1];   // (N,32) f32
    const int*   bid  = (const int*)d_in[2];     // (N,) i32
    const int N = in_sizes[2];

    float* out    = (float*)d_out;
    float* masked = out;                                   // 3*N floats
    float* sums   = out + (long)3 * N;                     // BV*32 floats
    float* counts = sums + (long)BATCHES * VOXELS * CH;    // BV floats
    float* inst   = counts + (long)BATCHES * VOXELS;       // 8 floats

    // --- 1) zero accumulators (sums + counts + inst): region is 16B aligned,
    //        element count divisible by 4.
    long accum_floats = (long)BATCHES * VOXELS * CH + (long)BATCHES * VOXELS + BATCHES;
    long n4 = accum_floats / 4;
    zero_accum<<<4096, 256, 0, stream>>>(sums, n4);

    // --- 2) scatter: one wave32 per point.
    long sthreads = (long)N * 32;
    int  sblocks  = (int)((sthreads + 255) / 256);
    scatter_points<<<sblocks, 256, 0, stream>>>(pc, feat, bid, masked, sums, counts, inst, N);

    // --- 3) finalize: one wave32 per voxel.
    long BV       = (long)BATCHES * VOXELS;
    long fthreads = BV * 32;
    int  fblocks  = (int)((fthreads + 255) / 256);
    finalize_voxels<<<fblocks, 256, 0, stream>>>(sums, counts, BV);
}